// SDPAttention_17179869508
// MI455X (gfx1250) — compile-verified
//
#include <hip/hip_runtime.h>
#include <hip/hip_bf16.h>

typedef __attribute__((ext_vector_type(2))) float v2f;
typedef __attribute__((ext_vector_type(8))) float v8f;

#define B_   16
#define T_   2048
#define D_   64
#define SSTR 2052          // padded LDS row stride (floats): 2052 % 64 == 4 -> conflict-free
#define NEGV (-1e9f)
#define SCALE 0.125f       // 1/sqrt(64)

// One block = 16 query rows of one batch. 128 threads = 4 waves.
// Phase 1: S = scale * Q K^T (+ causal mask) -> LDS via v_wmma_f32_16x16x4_f32
// Phase 2-4: row softmax in LDS; write normalized P to global sm_qk (single pass)
// Phase 5: O = P V via v_wmma_f32_16x16x4_f32
__global__ __launch_bounds__(128) void sdpa_fwd_kernel(const float* __restrict__ q,
                                                       const float* __restrict__ k,
                                                       const float* __restrict__ v,
                                                       float* __restrict__ outp,
                                                       float* __restrict__ smqk) {
    extern __shared__ float sS[];               // [16][SSTR] scores -> probs
    __shared__ float red[16 * 8];               // partial max/sum
    __shared__ float rowmax[16];
    __shared__ float rowsum[16];

    const int batch = blockIdx.y;
    const int m0    = blockIdx.x * 16;          // query row base
    const int tid   = threadIdx.x;
    const int lane  = tid & 31;
    const int wv    = tid >> 5;                 // wave 0..3
    const int half  = lane >> 4;                // 0 or 1 (K-split halves of the wave)
    const int l16   = lane & 15;
    const int mbase = half * 8;                 // C/D row base for this half-wave

    // ---------------- Phase 1: QK^T ----------------
    // A (Q tile, 16x4 slices): lane l16 = row, float2 at col kk*4 + 2*half
    v2f aq[16];
    {
        const float* qrow = q + ((size_t)batch * T_ + m0 + l16) * D_ + 2 * half;
#pragma unroll
        for (int kk = 0; kk < 16; ++kk) aq[kk] = *(const v2f*)(qrow + kk * 4);
    }

    // 128 key tiles of 16, interleaved across the 4 waves for load balance
    for (int j = wv; j < T_ / 16; j += 4) {
        const int nb = j * 16;
        if (nb > m0 + 15) {
            // fully masked tile: no compute, just fill NEG
#pragma unroll
            for (int r = 0; r < 8; ++r)
                sS[(mbase + r) * SSTR + nb + l16] = NEGV;
        } else {
            const float* krow = k + ((size_t)batch * T_ + nb + l16) * D_ + 2 * half;
            v8f acc = {};
#pragma unroll
            for (int kk = 0; kk < 16; ++kk) {
                v2f bk = *(const v2f*)(krow + kk * 4);
                acc = __builtin_amdgcn_wmma_f32_16x16x4_f32(
                    false, aq[kk], false, bk, (short)0, acc, false, false);
            }
            const int n = nb + l16;
#pragma unroll
            for (int r = 0; r < 8; ++r) {
                float s = acc[r] * SCALE;
                if (n > m0 + mbase + r) s = NEGV;   // causal mask
                sS[(mbase + r) * SSTR + n] = s;
            }
        }
    }
    __syncthreads();

    // ---------------- Phase 2: row max ----------------
    const int row   = tid & 15;
    const int chunk = tid >> 4;                 // 0..7, each covers 256 cols
    float* srow = sS + row * SSTR + chunk * 256;
    {
        float mx = -3.4e38f;
#pragma unroll 8
        for (int c = 0; c < 256; ++c) mx = fmaxf(mx, srow[c]);
        red[row * 8 + chunk] = mx;
    }
    __syncthreads();
    if (tid < 16) {
        float mx = red[tid * 8];
#pragma unroll
        for (int j = 1; j < 8; ++j) mx = fmaxf(mx, red[tid * 8 + j]);
        rowmax[tid] = mx;
    }
    __syncthreads();

    // ---------------- Phase 3: exp + row sum ----------------
    {
        const float rm = rowmax[row];
        float sum = 0.f;
#pragma unroll 8
        for (int c = 0; c < 256; ++c) {
            float e = __expf(srow[c] - rm);
            srow[c] = e;
            sum += e;
        }
        red[row * 8 + chunk] = sum;
    }
    __syncthreads();
    if (tid < 16) {
        float s = red[tid * 8];
#pragma unroll
        for (int j = 1; j < 8; ++j) s += red[tid * 8 + j];
        rowsum[tid] = s;
    }
    __syncthreads();

    // ---------------- Phase 4: normalize; write P (float4) ----------------
    {
        const float rinv = 1.0f / rowsum[row];
        float* gp = smqk + ((size_t)batch * T_ + m0 + row) * T_ + chunk * 256;
        float4* s4 = (float4*)srow;
        float4* g4 = (float4*)gp;
#pragma unroll 4
        for (int j = 0; j < 64; ++j) {
            float4 p = s4[j];
            p.x *= rinv; p.y *= rinv; p.z *= rinv; p.w *= rinv;
            s4[j] = p;
            g4[j] = p;
        }
    }
    __syncthreads();

    // ---------------- Phase 5: O = P V ----------------
    // wave wv owns output cols [wv*16, wv*16+16). Causality: P==0 for key >= m0+16.
    {
        const int kmax = m0 + 16;
        const float* vbase = v + (size_t)batch * T_ * D_ + wv * 16 + l16;
        v8f acc = {};
#pragma unroll 4
        for (int kk = 0; kk < kmax; kk += 4) {
            const int off = kk + 2 * half;      // this half-wave's K rows: off, off+1
            v2f ap = *(const v2f*)(sS + l16 * SSTR + off);   // P[row=l16][off..off+1]
            const float* vp = vbase + (size_t)off * D_;
            v2f bp;
            bp.x = vp[0];
            bp.y = vp[D_];
            acc = __builtin_amdgcn_wmma_f32_16x16x4_f32(
                false, ap, false, bp, (short)0, acc, false, false);
        }
#pragma unroll
        for (int r = 0; r < 8; ++r) {
            outp[((size_t)batch * T_ + m0 + mbase + r) * D_ + wv * 16 + l16] = acc[r];
        }
    }
}

extern "C" void kernel_launch(void* const* d_in, const int* in_sizes, int n_in,
                              void* d_out, int out_size, void* d_ws, size_t ws_size,
                              hipStream_t stream) {
    const float* q = (const float*)d_in[0];
    const float* k = (const float*)d_in[1];
    const float* v = (const float*)d_in[2];

    float* outp = (float*)d_out;                                  // [16,2048,64]
    float* smqk = outp + (size_t)B_ * T_ * D_;                    // [16,2048,2048]

    dim3 grid(T_ / 16, B_);   // 128 query tiles x 16 batches
    dim3 block(128);          // 4 waves
    size_t shmem = (size_t)16 * SSTR * sizeof(float);             // ~128.3 KB scores

    sdpa_fwd_kernel<<<grid, block, shmem, stream>>>(q, k, v, outp, smqk);
}